// FoundationalTimeSeriesModelV5_59983513256430
// MI455X (gfx1250) — compile-verified
//
#include <hip/hip_runtime.h>
#include <math.h>

// ---------------- model constants ----------------
#define Bq    32
#define Lq    512
#define Sq    32
#define PROJq 32
#define DMq   64
#define EPSF  1e-5f

// sinusoidal PE decay: log(10000)/PROJ
#define PE_DECAY 0.28782313662425575f

// chunked conv tiling: 128 output cols + 48 halo = 176 local cols = 11 tiles of 16
#define CCOLS 176
#define CHUNK 128
#define HALO  48

// transposed conv-weight LDS stride: 2*48 = 96 = 32 (mod 64) banks => the two
// lane-halves (ciA vs ciA+2) land on disjoint bank sets
#define CWS 48

typedef float v2f __attribute__((ext_vector_type(2)));
typedef float v8f __attribute__((ext_vector_type(8)));

// ---------------- workspace layout (floats) ----------------
#define WS_MEAN  0                         // 1024
#define WS_STD   1024                      // 1024
#define WS_LASTN 2048                      // 1024
#define WS_POOL  3072                      // 1024*32
#define WS_STFL  35840                     // 1024*32
#define WS_CTX   68608                     // 32*32*64
#define WS_GLOB  134144                    // 32*64
// total 136192 floats = 545 KB

// =====================================================================
// Kernel 1: RevIN + pointwise encoder + PE + 3 dilated causal convs
//           (WMMA f32 16x16x4, LDS-resident, chunked) + channel-LN +
//           mean-pool + stf_last.  One block per (b,s) sequence.
// =====================================================================
__global__ __launch_bounds__(256) void k_encoder(
    const float* __restrict__ x,      // (B,L,S)
    const float* __restrict__ smask,  // (B,S)
    const float* __restrict__ revw,   // (S)
    const float* __restrict__ revb,   // (S)
    const float* __restrict__ ew,     // (PROJ)
    const float* __restrict__ eb,     // (PROJ)
    const float* __restrict__ convw,  // (3,32,32,3)
    const float* __restrict__ convb,  // (3,32)
    const float* __restrict__ lng,    // (32)
    const float* __restrict__ lnb,    // (32)
    float* __restrict__ ws)
{
    __shared__ float bufA[32 * CCOLS];       // 5632 f
    __shared__ float bufB[32 * CCOLS];       // 5632 f
    __shared__ float cwT[96 * CWS];          // 4608 f: [(k*32+ci)*48 + co]
    __shared__ float red[256];
    __shared__ float pool_s[32];
    __shared__ float pefreq[16];
    __shared__ float s_mean, s_inv;

    const int seq  = blockIdx.x;            // b*32 + s
    const int b    = seq >> 5;
    const int s    = seq & 31;
    const int tid  = threadIdx.x;
    const int lane = tid & 31;
    const int wave = tid >> 5;
    const int m    = lane & 15;             // A row / B col / D col
    const int hi   = lane >> 4;             // K-half selector

    const float msk = smask[b * Sq + s];
    const float rw  = revw[s];
    const float rb  = revb[s];

    if (tid < 16) pefreq[tid] = expf(-PE_DECAY * (float)(2 * tid));

    // ---- RevIN stats over L (masked) ----
    float s1 = 0.f, s2 = 0.f;
    for (int l = tid; l < Lq; l += 256) {
        float v = x[(b * Lq + l) * Sq + s] * msk;
        s1 += v; s2 += v * v;
    }
    red[tid] = s1; __syncthreads();
    for (int st = 128; st > 0; st >>= 1) { if (tid < st) red[tid] += red[tid + st]; __syncthreads(); }
    if (tid == 0) s_mean = red[0] * (1.f / (float)Lq);
    __syncthreads();
    red[tid] = s2; __syncthreads();
    for (int st = 128; st > 0; st >>= 1) { if (tid < st) red[tid] += red[tid + st]; __syncthreads(); }
    if (tid == 0) {
        float mn  = s_mean;
        float var = red[0] * (1.f / (float)Lq) - mn * mn;
        float sd  = sqrtf(var + EPSF);
        s_inv = 1.f / sd;
        ws[WS_MEAN + seq] = mn;
        ws[WS_STD  + seq] = sd;
    }
    if (tid < 32) pool_s[tid] = 0.f;
    __syncthreads();

    float pacc[32];
    #pragma unroll
    for (int c = 0; c < 32; ++c) pacc[c] = 0.f;

    // ---- process L in 4 chunks of 128 with 48 halo ----
    for (int chunk = 0; chunk < 4; ++chunk) {
        const int c0       = chunk * CHUNK;
        const int base_abs = c0 - HALO;

        // -- fill h = xn*ew + eb + PE into bufA (pointwise, no halo cost) --
        if (tid < CCOLS) {
            int abs_l = base_abs + tid;
            if (abs_l < 0) {
                #pragma unroll
                for (int p = 0; p < 32; ++p) bufA[p * CCOLS + tid] = 0.f;
            } else {
                float xv = x[(b * Lq + abs_l) * Sq + s] * msk;
                float xn = (xv - s_mean) * s_inv * rw + rb;
                if (abs_l == Lq - 1) ws[WS_LASTN + seq] = xn;
                #pragma unroll
                for (int i = 0; i < 16; ++i) {
                    float ang = (float)abs_l * pefreq[i];
                    bufA[(2 * i)     * CCOLS + tid] = xn * ew[2 * i]     + eb[2 * i]     + sinf(ang);
                    bufA[(2 * i + 1) * CCOLS + tid] = xn * ew[2 * i + 1] + eb[2 * i + 1] + cosf(ang);
                }
            }
        }
        __syncthreads();

        // -- 3 dilated causal conv layers via WMMA f32 16x16x4 --
        float* pin  = bufA;
        float* pout = bufB;
        for (int layer = 0; layer < 3; ++layer) {
            const int    dil    = 1 << layer;
            const float* cw     = convw + layer * 32 * 32 * 3;
            const float* cb     = convb + layer * 32;
            const int    jstart = 1 + layer;          // valid l-tiles shrink by 1/layer
            const int    ntiles = (10 - layer) * 2;   // l-tiles * 2 co-tiles

            // stage this layer's weights transposed into LDS: cwT[(k*32+ci)*CWS + co]
            for (int e = tid; e < 96 * 32; e += 256) {
                int row = e >> 5;                  // k*32 + ci
                int co  = e & 31;
                int k   = row >> 5;
                int ci  = row & 31;
                cwT[row * CWS + co] = cw[co * 96 + ci * 3 + k];
            }
            __syncthreads();

            for (int t = wave; t < ntiles; t += 8) {
                const int jt      = jstart + (t >> 1);
                const int co_base = (t & 1) << 4;
                const int lloc0   = jt << 4;
                const int labs0   = base_abs + lloc0;   // multiple of 16 (tile-uniform sign)

                v8f acc = {0.f, 0.f, 0.f, 0.f, 0.f, 0.f, 0.f, 0.f};
                #pragma unroll
                for (int ci0 = 0; ci0 < 32; ci0 += 4) {
                    const int ciA = ci0 + 2 * hi;     // this lane's K-pair of input channels
                    #pragma unroll
                    for (int k = 0; k < 3; ++k) {
                        const int back = (2 - k) * dil;
                        const int lidx = lloc0 + m - back;   // >= 14 always, in-buffer
                        v2f a, bm;
                        // A (16 co x 4 ci): lanes0-15 -> K0,K1 ; lanes16-31 -> K2,K3
                        a[0] = cwT[(k * 32 + ciA)     * CWS + co_base + m];
                        a[1] = cwT[(k * 32 + ciA + 1) * CWS + co_base + m];
                        // B (4 ci x 16 l): buffer is zero for abs<0 -> unconditional
                        bm[0] = pin[ciA       * CCOLS + lidx];
                        bm[1] = pin[(ciA + 1) * CCOLS + lidx];
                        acc = __builtin_amdgcn_wmma_f32_16x16x4_f32(
                                  false, a, false, bm, (short)0, acc, false, false);
                    }
                }
                // D (16x16): VGPR r -> rows r (lanes0-15) / r+8 (lanes16-31)
                // causal padding: tiles entirely at abs<0 must store exact zeros
                #pragma unroll
                for (int r = 0; r < 8; ++r) {
                    const int co = co_base + (hi << 3) + r;
                    float v = fmaxf(acc[r] + cb[co], 0.f);
                    if (labs0 < 0) v = 0.f;
                    pout[co * CCOLS + lloc0 + m] = v;
                }
            }
            __syncthreads();
            float* tsw = pin; pin = pout; pout = tsw;
        }
        // after 3 swaps, final conv output is in `pin`

        // -- channel LayerNorm + mask + pooling over the 128 real cols --
        if (tid < CHUNK) {
            const int col   = HALO + tid;
            const int abs_l = c0 + tid;
            float v[32];
            float mn = 0.f;
            #pragma unroll
            for (int c = 0; c < 32; ++c) { v[c] = pin[c * CCOLS + col]; mn += v[c]; }
            mn *= (1.f / 32.f);
            float var = 0.f;
            #pragma unroll
            for (int c = 0; c < 32; ++c) { float d = v[c] - mn; var += d * d; }
            var *= (1.f / 32.f);
            float inv = rsqrtf(var + EPSF);
            #pragma unroll
            for (int c = 0; c < 32; ++c) {
                float st = ((v[c] - mn) * inv * lng[c] + lnb[c]) * msk;
                pacc[c] += st;
                if (abs_l == Lq - 1) ws[WS_STFL + seq * 32 + c] = st;
            }
        }
        __syncthreads();
    }

    if (tid < CHUNK)
        for (int c = 0; c < 32; ++c) atomicAdd(&pool_s[c], pacc[c]);
    __syncthreads();
    if (tid < 32)
        ws[WS_POOL + seq * 32 + tid] = pool_s[tid] * (1.f / (float)Lq);
}

// =====================================================================
// Kernel 2: pool_proj + pos, 2-layer transformer over S=32 tokens,
//           final LN, ctx and glob.  One block per batch element.
// =====================================================================
__device__ inline void ln_row(const float* in, float* out,
                              const float* g, const float* bsh, int n)
{
    float mn = 0.f;
    for (int i = 0; i < n; ++i) mn += in[i];
    mn /= (float)n;
    float var = 0.f;
    for (int i = 0; i < n; ++i) { float d = in[i] - mn; var += d * d; }
    var /= (float)n;
    float inv = rsqrtf(var + EPSF);
    for (int i = 0; i < n; ++i) out[i] = (in[i] - mn) * inv * g[i] + bsh[i];
}

__global__ __launch_bounds__(256) void k_transformer(
    const float* __restrict__ smask,
    const float* __restrict__ ppw,  const float* __restrict__ ppb,
    const float* __restrict__ pos,
    const float* __restrict__ ln1g, const float* __restrict__ ln1b,
    const float* __restrict__ qkvw, const float* __restrict__ qkvb,
    const float* __restrict__ outw, const float* __restrict__ outb,
    const float* __restrict__ ln2g, const float* __restrict__ ln2b,
    const float* __restrict__ ff1w, const float* __restrict__ ff1b,
    const float* __restrict__ ff2w, const float* __restrict__ ff2b,
    const float* __restrict__ nrmg, const float* __restrict__ nrmb,
    float* __restrict__ ws)
{
    __shared__ float t[32 * 64];
    __shared__ float hh[32 * 64];
    __shared__ float qkv[32 * 192];
    __shared__ float scratch[4 * 32 * 32];   // att scores / ff1 hidden

    const int b   = blockIdx.x;
    const int tid = threadIdx.x;
    const float* pool = ws + WS_POOL + b * 32 * 32;

    // t = pooled @ ppw.T + ppb + pos
    for (int e = tid; e < 32 * 64; e += 256) {
        int r = e >> 6, dm = e & 63;
        float acc = ppb[dm];
        for (int c = 0; c < 32; ++c) acc += pool[r * 32 + c] * ppw[dm * 32 + c];
        t[e] = acc + pos[r * 64 + dm];
    }
    __syncthreads();

    for (int l = 0; l < 2; ++l) {
        // LN1 -> hh
        if (tid < 32) ln_row(t + tid * 64, hh + tid * 64, ln1g + l * 64, ln1b + l * 64, 64);
        __syncthreads();
        // qkv = hh @ qkvw.T + b
        for (int e = tid; e < 32 * 192; e += 256) {
            int r = e / 192, j = e % 192;
            float acc = qkvb[l * 192 + j];
            const float* wr = qkvw + (l * 192 + j) * 64;
            for (int d = 0; d < 64; ++d) acc += hh[r * 64 + d] * wr[d];
            qkv[e] = acc;
        }
        __syncthreads();
        // scores + mask + softmax  (thread = (head, q))
        if (tid < 128) {
            int h = tid >> 5, q = tid & 31;
            float* row = scratch + (h * 32 + q) * 32;
            float mx = -1e30f;
            for (int kk = 0; kk < 32; ++kk) {
                float acc = 0.f;
                for (int d = 0; d < 16; ++d)
                    acc += qkv[q * 192 + h * 16 + d] * qkv[kk * 192 + 64 + h * 16 + d];
                acc *= 0.25f;                         // 1/sqrt(16)
                if (smask[b * 32 + kk] == 0.f) acc = -1e9f;
                row[kk] = acc; mx = fmaxf(mx, acc);
            }
            float sm = 0.f;
            for (int kk = 0; kk < 32; ++kk) { float e2 = expf(row[kk] - mx); row[kk] = e2; sm += e2; }
            float inv = 1.f / sm;
            for (int kk = 0; kk < 32; ++kk) row[kk] *= inv;
        }
        __syncthreads();
        // o = att @ v  -> hh
        for (int e = tid; e < 32 * 64; e += 256) {
            int q = e >> 6, hd = e & 63, h = hd >> 4;
            const float* ar = scratch + (h * 32 + q) * 32;
            float acc = 0.f;
            for (int kk = 0; kk < 32; ++kk) acc += ar[kk] * qkv[kk * 192 + 128 + hd];
            hh[e] = acc;
        }
        __syncthreads();
        // t += o @ outw.T + outb
        for (int e = tid; e < 32 * 64; e += 256) {
            int r = e >> 6, dm = e & 63;
            float acc = outb[l * 64 + dm];
            const float* wr = outw + (l * 64 + dm) * 64;
            for (int j = 0; j < 64; ++j) acc += hh[r * 64 + j] * wr[j];
            t[e] += acc;
        }
        __syncthreads();
        // LN2 -> hh
        if (tid < 32) ln_row(t + tid * 64, hh + tid * 64, ln2g + l * 64, ln2b + l * 64, 64);
        __syncthreads();
        // ff1 (relu) -> scratch (32x128)
        for (int e = tid; e < 32 * 128; e += 256) {
            int r = e >> 7, j = e & 127;
            float acc = ff1b[l * 128 + j];
            const float* wr = ff1w + (l * 128 + j) * 64;
            for (int d = 0; d < 64; ++d) acc += hh[r * 64 + d] * wr[d];
            scratch[e] = fmaxf(acc, 0.f);
        }
        __syncthreads();
        // t += ff1 @ ff2w.T + b
        for (int e = tid; e < 32 * 64; e += 256) {
            int r = e >> 6, dm = e & 63;
            float acc = ff2b[l * 64 + dm];
            const float* wr = ff2w + (l * 64 + dm) * 128;
            for (int j = 0; j < 128; ++j) acc += scratch[r * 128 + j] * wr[j];
            t[e] += acc;
        }
        __syncthreads();
    }

    // final LN * mask -> ctx ; hh holds ctx*mask for glob numerator
    if (tid < 32) {
        float tmp[64];
        ln_row(t + tid * 64, tmp, nrmg, nrmb, 64);
        float mv = smask[b * 32 + tid];
        for (int d = 0; d < 64; ++d) {
            float cv = tmp[d] * mv;
            ws[WS_CTX + (b * 32 + tid) * 64 + d] = cv;
            hh[tid * 64 + d] = cv * mv;
        }
    }
    __syncthreads();
    if (tid < 64) {
        float den = 0.f;
        for (int r = 0; r < 32; ++r) den += smask[b * 32 + r];
        den = fmaxf(den, 1.f);
        float acc = 0.f;
        for (int r = 0; r < 32; ++r) acc += hh[r * 64 + tid];
        ws[WS_GLOB + b * 64 + tid] = acc / den;
    }
}

// =====================================================================
// Kernel 3: 3x top-1 MoE + heads (pred w/ RevIN denorm, fail, rca).
//           One block per batch element.
// =====================================================================
__global__ __launch_bounds__(256) void k_moe_heads(
    const float* __restrict__ revw, const float* __restrict__ revb,
    const float* __restrict__ gatew, const float* __restrict__ gateb,
    const float* __restrict__ ew1, const float* __restrict__ eb1,
    const float* __restrict__ ew2, const float* __restrict__ eb2,
    const float* __restrict__ predw, const float* __restrict__ predb,
    const float* __restrict__ failw, const float* __restrict__ failb,
    const float* __restrict__ rcaw,  const float* __restrict__ rcab,
    const float* __restrict__ ws, float* __restrict__ outp)
{
    __shared__ float g[64];
    __shared__ float moe[3][64];
    __shared__ float h1[128];
    __shared__ float logit[8];
    __shared__ int   sel;

    const int b = blockIdx.x, tid = threadIdx.x;
    if (tid < 64) g[tid] = ws[WS_GLOB + b * 64 + tid];
    __syncthreads();

    for (int hd = 0; hd < 3; ++hd) {
        if (tid < 8) {
            float acc = gateb[hd * 8 + tid];
            const float* wr = gatew + (hd * 8 + tid) * 64;
            for (int d = 0; d < 64; ++d) acc += g[d] * wr[d];
            logit[tid] = acc;
        }
        __syncthreads();
        if (tid == 0) {
            int bi = 0; float bv = logit[0];
            for (int e = 1; e < 8; ++e) if (logit[e] > bv) { bv = logit[e]; bi = e; }
            sel = bi;
        }
        __syncthreads();
        const int eidx = hd * 8 + sel;
        if (tid < 128) {
            float acc = eb1[eidx * 128 + tid];
            const float* wr = ew1 + (eidx * 128 + tid) * 64;
            for (int d = 0; d < 64; ++d) acc += g[d] * wr[d];
            h1[tid] = fmaxf(acc, 0.f);
        }
        __syncthreads();
        if (tid < 64) {
            float acc = eb2[eidx * 64 + tid];
            const float* wr = ew2 + (eidx * 64 + tid) * 128;
            for (int j = 0; j < 128; ++j) acc += h1[j] * wr[j];
            moe[hd][tid] = acc;
        }
        __syncthreads();
    }

    // fail head: (B,3)
    if (tid < 3) {
        float acc = failb[tid];
        for (int d = 0; d < 64; ++d) acc += moe[1][d] * failw[tid * 64 + d];
        outp[Bq * Sq * 8 + b * 3 + tid] = acc;
    }

    // pred head: (B,S,8) with RevIN de-normalization  (256 = 32*8 tasks)
    {
        const int s = tid >> 3, ph = tid & 7;
        const int seq = b * 32 + s;
        const float* stfl = ws + WS_STFL + seq * 32;
        const float* ctx  = ws + WS_CTX  + seq * 64;
        const float* wr   = predw + ph * 160;
        float acc = predb[ph];
        for (int c = 0; c < 32; ++c) acc += stfl[c]   * wr[c];
        for (int c = 0; c < 64; ++c) acc += ctx[c]    * wr[32 + c];
        for (int c = 0; c < 64; ++c) acc += moe[0][c] * wr[96 + c];
        float pn = ws[WS_LASTN + seq] + acc;
        float pr = (pn - revb[s]) / (revw[s] + EPSF) * ws[WS_STD + seq] + ws[WS_MEAN + seq];
        outp[seq * 8 + ph] = pr;
    }

    // rca head: (B,S)
    if (tid < 32) {
        const int seq = b * 32 + tid;
        const float* stfl = ws + WS_STFL + seq * 32;
        const float* ctx  = ws + WS_CTX  + seq * 64;
        float acc = rcab[0];
        for (int c = 0; c < 32; ++c) acc += stfl[c]   * rcaw[c];
        for (int c = 0; c < 64; ++c) acc += ctx[c]    * rcaw[32 + c];
        for (int c = 0; c < 64; ++c) acc += moe[2][c] * rcaw[96 + c];
        outp[Bq * Sq * 8 + Bq * 3 + seq] = acc;
    }
}

// =====================================================================
extern "C" void kernel_launch(void* const* d_in, const int* in_sizes, int n_in,
                              void* d_out, int out_size, void* d_ws, size_t ws_size,
                              hipStream_t stream)
{
    (void)in_sizes; (void)n_in; (void)out_size; (void)ws_size;
    const float* x     = (const float*)d_in[0];
    const float* smask = (const float*)d_in[1];
    const float* revw  = (const float*)d_in[2];
    const float* revb  = (const float*)d_in[3];
    const float* ew    = (const float*)d_in[4];
    const float* eb    = (const float*)d_in[5];
    const float* convw = (const float*)d_in[6];
    const float* convb = (const float*)d_in[7];
    const float* lng   = (const float*)d_in[8];
    const float* lnb   = (const float*)d_in[9];
    const float* ppw   = (const float*)d_in[10];
    const float* ppb   = (const float*)d_in[11];
    const float* pos   = (const float*)d_in[12];
    const float* ln1g  = (const float*)d_in[13];
    const float* ln1b  = (const float*)d_in[14];
    const float* qkvw  = (const float*)d_in[15];
    const float* qkvb  = (const float*)d_in[16];
    const float* outw  = (const float*)d_in[17];
    const float* outb  = (const float*)d_in[18];
    const float* ln2g  = (const float*)d_in[19];
    const float* ln2b  = (const float*)d_in[20];
    const float* ff1w  = (const float*)d_in[21];
    const float* ff1b  = (const float*)d_in[22];
    const float* ff2w  = (const float*)d_in[23];
    const float* ff2b  = (const float*)d_in[24];
    const float* nrmg  = (const float*)d_in[25];
    const float* nrmb  = (const float*)d_in[26];
    const float* gatew = (const float*)d_in[27];
    const float* gateb = (const float*)d_in[28];
    const float* ew1   = (const float*)d_in[29];
    const float* eb1   = (const float*)d_in[30];
    const float* ew2   = (const float*)d_in[31];
    const float* eb2   = (const float*)d_in[32];
    const float* predw = (const float*)d_in[33];
    const float* predb = (const float*)d_in[34];
    const float* failw = (const float*)d_in[35];
    const float* failb = (const float*)d_in[36];
    const float* rcaw  = (const float*)d_in[37];
    const float* rcab  = (const float*)d_in[38];

    float* ws   = (float*)d_ws;
    float* outp = (float*)d_out;

    k_encoder<<<Bq * Sq, 256, 0, stream>>>(x, smask, revw, revb, ew, eb,
                                           convw, convb, lng, lnb, ws);
    k_transformer<<<Bq, 256, 0, stream>>>(smask, ppw, ppb, pos,
                                          ln1g, ln1b, qkvw, qkvb, outw, outb,
                                          ln2g, ln2b, ff1w, ff1b, ff2w, ff2b,
                                          nrmg, nrmb, ws);
    k_moe_heads<<<Bq, 256, 0, stream>>>(revw, revb, gatew, gateb,
                                        ew1, eb1, ew2, eb2,
                                        predw, predb, failw, failb,
                                        rcaw, rcab, ws, outp);
}